// MVCNNWithSparseConv_46076409152321
// MI455X (gfx1250) — compile-verified
//
#include <hip/hip_runtime.h>
#include <hip/hip_bf16.h>

// Problem constants (match reference setup_inputs()):
#define BATCH   4
#define NPTS    65536
#define CHN     256
#define GRID_G  8
#define NVOX    512            // 8^3
#define CHUNK   2048           // index ints per LDS chunk (8 KB)
#define NCHUNK  (NPTS / CHUNK) // 32

#if defined(__gfx1250__) && \
    __has_builtin(__builtin_amdgcn_global_load_async_to_lds_b128) && \
    __has_builtin(__builtin_amdgcn_s_wait_asynccnt)
#define USE_ASYNC 1
#else
#define USE_ASYNC 0
#endif

typedef __attribute__((ext_vector_type(4))) int v4i;
typedef __attribute__((address_space(1))) v4i gv4i;   // global
typedef __attribute__((address_space(3))) v4i lv4i;   // LDS

// ---------------------------------------------------------------------------
// Kernel 1: per-point voxel index, matching
//   jnp.clip(((pc + 1.0) * 0.5 * (G - 1e-5)).astype(int32), 0, G-1)
// ---------------------------------------------------------------------------
__global__ __launch_bounds__(256) void voxel_idx_kernel(
    const float* __restrict__ pc, int* __restrict__ vidx, int total) {
  int i = blockIdx.x * blockDim.x + threadIdx.x;
  if (i >= total) return;
  const float s = 8.0f - 1e-5f;   // G - 1e-5 as f32
  float x = pc[3 * i + 0];
  float y = pc[3 * i + 1];
  float z = pc[3 * i + 2];
  // same op order as reference: (p + 1.0) * 0.5, then * (G - 1e-5), trunc
  int gx = (int)(((x + 1.0f) * 0.5f) * s);
  int gy = (int)(((y + 1.0f) * 0.5f) * s);
  int gz = (int)(((z + 1.0f) * 0.5f) * s);
  gx = gx < 0 ? 0 : (gx > 7 ? 7 : gx);
  gy = gy < 0 ? 0 : (gy > 7 ? 7 : gy);
  gz = gz < 0 ? 0 : (gz > 7 ? 7 : gz);
  vidx[i] = gx * 64 + gy * 8 + gz;
}

// ---------------------------------------------------------------------------
// Async double-buffered chunk prefetch: 2048 ints (8 KB) per chunk.
// 256 threads x 2 x b128 (16 B per lane per op) => 2 outstanding per wave.
// ---------------------------------------------------------------------------
__device__ __forceinline__ void prefetch_chunk(const int* __restrict__ gsrc,
                                               int* lds_dst, int tid) {
#if USE_ASYNC
#pragma unroll
  for (int j = 0; j < 2; ++j) {
    int e = (j * 256 + tid) * 4;   // int index; 16-byte granules
    __builtin_amdgcn_global_load_async_to_lds_b128(
        (gv4i*)(gsrc + e), (lv4i*)(lds_dst + e), 0, 0);
  }
#else
#pragma unroll
  for (int j = 0; j < 2; ++j) {
    int e = j * 256 + tid;         // int4 index
    reinterpret_cast<int4*>(lds_dst)[e] =
        reinterpret_cast<const int4*>(gsrc)[e];
  }
#endif
}

// ---------------------------------------------------------------------------
// Kernel 2: one workgroup per (batch, voxel). Stream indices through LDS,
// build a match queue with LDS atomics, gather matching feature rows
// (thread == channel, fully coalesced 1 KB rows), normalize by count.
// ---------------------------------------------------------------------------
__global__ __launch_bounds__(256) void voxel_gather_kernel(
    const int* __restrict__ vidx, const float* __restrict__ feats,
    float* __restrict__ out) {
  __shared__ __attribute__((aligned(16))) int s_idx[2][CHUNK];
  __shared__ int s_queue[CHUNK];
  __shared__ int s_qcount;

  const int tid = threadIdx.x;
  const int b   = blockIdx.x >> 9;          // / NVOX
  const int v   = blockIdx.x & (NVOX - 1);  // % NVOX

  const int*   idxBase  = vidx  + (size_t)b * NPTS;
  const float* featBase = feats + (size_t)b * NPTS * CHN;

  float acc = 0.0f;
  int   cnt = 0;

  // warm up buffer 0 with chunk 0
  prefetch_chunk(idxBase, &s_idx[0][0], tid);

  for (int c = 0; c < NCHUNK; ++c) {
    const int  cur  = c & 1;
    const bool more = (c + 1 < NCHUNK);

    if (tid == 0) s_qcount = 0;
    if (more)
      prefetch_chunk(idxBase + (c + 1) * CHUNK, &s_idx[cur ^ 1][0], tid);

#if USE_ASYNC
    if (more) {
      __builtin_amdgcn_s_wait_asynccnt(2);  // newest 2 (next chunk) may fly
    } else {
      __builtin_amdgcn_s_wait_asynccnt(0);
    }
#endif
    __syncthreads();  // chunk c resident in LDS, qcount reset visible

    // scan this chunk: 8 indices per thread, conflict-free LDS reads
#pragma unroll
    for (int k = 0; k < CHUNK / 256; ++k) {
      int p = k * 256 + tid;
      if (s_idx[cur][p] == v) {
        int pos = atomicAdd(&s_qcount, 1);   // ds_add_rtn_u32
        s_queue[pos] = c * CHUNK + p;
      }
    }
    __syncthreads();  // queue complete

    const int qn = s_qcount;
    for (int q = 0; q < qn; ++q) {
      int p = s_queue[q];  // LDS broadcast
      acc += featBase[(size_t)p * CHN + tid];  // 1 KB contiguous per row
    }
    cnt += qn;
    __syncthreads();  // queue consumed before next iteration resets qcount
  }

  float denom = (cnt > 0) ? (float)cnt : 1.0f;
  out[((size_t)b * NVOX + v) * CHN + tid] = acc / denom;
}

// ---------------------------------------------------------------------------
// Launch
// ---------------------------------------------------------------------------
extern "C" void kernel_launch(void* const* d_in, const int* in_sizes, int n_in,
                              void* d_out, int out_size, void* d_ws, size_t ws_size,
                              hipStream_t stream) {
  const float* pc    = (const float*)d_in[0];  // (B, N, 3) f32
  const float* feats = (const float*)d_in[1];  // (B, N, C) f32
  float*       out   = (float*)d_out;          // (B, V, C) f32
  int*         vidx  = (int*)d_ws;             // B*N ints = 1 MB scratch

  const int total = BATCH * NPTS;

  voxel_idx_kernel<<<total / 256, 256, 0, stream>>>(pc, vidx, total);
  voxel_gather_kernel<<<BATCH * NVOX, 256, 0, stream>>>(vidx, feats, out);
  (void)in_sizes; (void)n_in; (void)out_size; (void)ws_size;
}